// PlasticLinear_85512798863597
// MI455X (gfx1250) — compile-verified
//
#include <hip/hip_runtime.h>

// Problem constants (fixed by the reference harness)
#define B_   32
#define T_   256
#define DIN  512
#define DOUT 512
#define ETA_C 0.1f
#define ROWS 128          // Dout rows owned by one phase-2 workgroup (4 slices)

typedef float v2f __attribute__((ext_vector_type(2)));
typedef float v8f __attribute__((ext_vector_type(8)));

// ---------------------------------------------------------------------------
// Phase 1: Z = X @ W^T + bias   (M=B*T=8192, N=DOUT=512, K=DIN=512), fp32 WMMA.
// One wave computes one 16x16 output tile, accumulating over K in steps of 4
// with V_WMMA_F32_16X16X4_F32. Exact fp32 arithmetic (matches reference).
//
// A-frag (16x4 f32): lanes 0-15 hold row M=l, K={0,1}; lanes 16-31 row M=l-16,
// K={2,3} -> per-lane contiguous float2 from the X row.
// B-frag mirrors A transposed: B[k][n] = W[n][k] -> per-lane float2 from W row n.
// C/D (16x16 f32, 8 VGPRs): VGPR j = row j (lanes 0-15) / row j+8 (lanes 16-31).
// ---------------------------------------------------------------------------
__global__ void __launch_bounds__(256)
plastic_gemm_f32(const float* __restrict__ X, const float* __restrict__ Wm,
                 const float* __restrict__ bias, float* __restrict__ Z) {
  const int M = B_ * T_, N = DOUT, K = DIN;
  const int lane = threadIdx.x & 31;
  const int wave = threadIdx.x >> 5;
  const int tile = blockIdx.x * 8 + wave;           // 8 waves per block
  const int ntn  = N >> 4;                          // 32 N-tiles
  const int tm   = tile / ntn;
  const int tn   = tile - tm * ntn;
  if (tm * 16 >= M) return;

  const int half = lane >> 4;                       // K-pair select
  const int l16  = lane & 15;
  const float* arow = X  + (size_t)(tm * 16 + l16) * K + half * 2;
  const float* brow = Wm + (size_t)(tn * 16 + l16) * K + half * 2;

  v8f acc = {};
#pragma unroll 4
  for (int k = 0; k < K; k += 4) {
    v2f a = *(const v2f*)(arow + k);
    v2f b = *(const v2f*)(brow + k);
    acc = __builtin_amdgcn_wmma_f32_16x16x4_f32(
        /*neg_a=*/false, a, /*neg_b=*/false, b,
        /*c_mod=*/(short)0, acc, /*reuse_a=*/false, /*reuse_b=*/false);
  }

  const float bc = bias[tn * 16 + l16];
  float* zp = Z + (size_t)(tm * 16 + half * 8) * N + tn * 16 + l16;
#pragma unroll
  for (int j = 0; j < 8; ++j) zp[(size_t)j * N] = acc[j] + bc;
}

// ---------------------------------------------------------------------------
// Phase 2: sequential Oja scan. Block (b, slice) owns H[b, slice*128 : +128, :]
// resident in LDS (128*512 fp32 = 256 KB, fits CDNA5's 320 KB/WGP) for the
// entire T=256 scan -> zero HBM traffic for the fast-weight state.
//
// Row-split over Dout => no cross-workgroup communication: each block computes
// y for its own rows (dot over full Din) and updates only its own rows.
// 2 threads per row; pair dot-reduction via wave32 shfl_xor so both threads of
// a row know y and the H update needs no extra barrier.
// ---------------------------------------------------------------------------
__global__ void __launch_bounds__(256, 1)
plastic_oja_scan(const float* __restrict__ x, const float* __restrict__ mods,
                 float* __restrict__ out /* in: Z, out: y */) {
  extern __shared__ float smem[];
  float* Hs = smem;                       // ROWS * DIN
  float* xs = smem + ROWS * DIN;          // DIN

  const int b     = blockIdx.x >> 2;
  const int slice = blockIdx.x & 3;
  const int o0    = slice * ROWS;
  const int tid   = threadIdx.x;

  // Zero the fast-weight state.
  for (int i = tid; i < ROWS * DIN; i += 256) Hs[i] = 0.0f;

  const int r = tid >> 1;                 // row 0..127
  const int h = tid & 1;                  // column-half 0/1
  float4*       Hrow = (float4*)(Hs + r * DIN + h * (DIN / 2));
  const float4* xh   = (const float4*)(xs + h * (DIN / 2));
  const int nvec = (DIN / 2) / 4;         // 64 float4 per thread

  const float* xb = x    + (size_t)b * T_ * DIN;
  const float* mb = mods + (size_t)b * T_;
  float*       ob = out  + (size_t)b * T_ * DOUT + o0;

  __syncthreads();

  for (int t = 0; t < T_; ++t) {
    // Stage x_t into LDS (128 float4 loads, threads 0..127).
    const float4* xg = (const float4*)(xb + (size_t)t * DIN);
    if (tid < DIN / 4) ((float4*)xs)[tid] = xg[tid];
    __syncthreads();

    // dot = H[r,:] . x_t  (each thread: half the row, vectorized)
    float dot = 0.0f;
#pragma unroll 8
    for (int j = 0; j < nvec; ++j) {
      float4 hv = Hrow[j];
      float4 xv = xh[j];
      dot = __builtin_fmaf(hv.x, xv.x, dot);
      dot = __builtin_fmaf(hv.y, xv.y, dot);
      dot = __builtin_fmaf(hv.z, xv.z, dot);
      dot = __builtin_fmaf(hv.w, xv.w, dot);
    }
    dot += __shfl_xor(dot, 1, 32);        // pair-reduce; both halves get full dot

    const float y = ob[(size_t)t * DOUT + r] + dot;   // Z (precomputed) + H.x
    if (h == 0) ob[(size_t)t * DOUT + r] = y;

    // Oja update: H = H * (1 - eta*m*y^2) + (eta*m*y) * x^T
    const float m   = mb[t];
    const float a   = ETA_C * m * y;
    const float dcy = 1.0f - a * y;
#pragma unroll 8
    for (int j = 0; j < nvec; ++j) {
      float4 hv = Hrow[j];
      float4 xv = xh[j];
      hv.x = __builtin_fmaf(hv.x, dcy, a * xv.x);
      hv.y = __builtin_fmaf(hv.y, dcy, a * xv.y);
      hv.z = __builtin_fmaf(hv.z, dcy, a * xv.z);
      hv.w = __builtin_fmaf(hv.w, dcy, a * xv.w);
      Hrow[j] = hv;
    }
    __syncthreads();                      // xs reused next iteration
  }
}

extern "C" void kernel_launch(void* const* d_in, const int* in_sizes, int n_in,
                              void* d_out, int out_size, void* d_ws, size_t ws_size,
                              hipStream_t stream) {
  const float* x    = (const float*)d_in[0];
  const float* W    = (const float*)d_in[1];
  const float* bias = (const float*)d_in[2];
  const float* mods = (const float*)d_in[3];
  float* out = (float*)d_out;

  // Phase 1: 16384 tiles (512 x 32), 8 waves/block -> 2048 blocks.
  plastic_gemm_f32<<<2048, 256, 0, stream>>>(x, W, bias, out);

  // Phase 2: 32 batches x 4 row-slices = 128 blocks, 256 KB + x buffer in LDS.
  const size_t lds = (size_t)(ROWS * DIN + DIN) * sizeof(float);
  (void)hipFuncSetAttribute((const void*)plastic_oja_scan,
                            hipFuncAttributeMaxDynamicSharedMemorySize, (int)lds);
  plastic_oja_scan<<<128, 256, lds, stream>>>(x, mods, out);
}